// MLPAttention_12223476924654
// MI455X (gfx1250) — compile-verified
//
#include <hip/hip_runtime.h>
#include <math.h>

// Problem dims (from reference setup_inputs): N=8, Tx=1024, Ty=128,
// enc_dim=512, dec_dim=1024, Ca=128.
#define BN    8
#define TX    1024
#define TY    128
#define ENC   512
#define DEC   1024
#define CA    128

typedef __attribute__((ext_vector_type(2))) float v2f;
typedef __attribute__((ext_vector_type(8))) float v8f;

// gfx1250 has a hardware V_TANH_F32 transcendental; use it when the builtin
// exists (it co-executes with VALU FMAs, unlike the ~15-op OCML polynomial).
__device__ __forceinline__ float hw_tanhf(float x) {
#if defined(__has_builtin)
#if __has_builtin(__builtin_amdgcn_tanhf)
  return __builtin_amdgcn_tanhf(x);
#else
  return tanhf(x);
#endif
#else
  return tanhf(x);
#endif
}

// C[M,N] = A[M,K] * B[K,N] (+ bias[N]), N must be 128 (8 waves x 16 cols).
// One wave computes a 16x16 tile with V_WMMA_F32_16X16X4_F32, K-loop step 4.
__global__ __launch_bounds__(256)
void gemm_wmma_f32_kernel(const float* __restrict__ A, const float* __restrict__ B,
                          const float* __restrict__ bias, float* __restrict__ C,
                          int M, int K, int N) {
  const int wave = threadIdx.x >> 5;          // 0..7 -> N tile
  const int lane = threadIdx.x & 31;
  const int m0   = blockIdx.x * 16;
  const int n0   = wave * 16;
  const int mrow = m0 + (lane & 15);
  const int koff = (lane >> 4) * 2;           // lanes 16..31 carry K+2,K+3
  const int col  = n0 + (lane & 15);

  // Pointer inductions with constant strides -> global_load offset addressing.
  const float* aptr = A + (size_t)mrow * K + koff;        // advances by 4
  const float* bptr = B + (size_t)koff * N + col;         // advances by 4*N

  v8f acc = {};
  #pragma unroll 4
  for (int k = 0; k < K; k += 4) {
    // A 16x4 tile: lane<16 -> {K=k,k+1}, lane>=16 -> {K=k+2,k+3}
    v2f a = *(const v2f*)aptr;
    // B 4x16 tile: same K split, column n0+(lane&15)
    v2f b;
    b[0] = bptr[0];
    b[1] = bptr[N];
    acc = __builtin_amdgcn_wmma_f32_16x16x4_f32(
        /*neg_a=*/false, a, /*neg_b=*/false, b,
        /*c_mod=*/(short)0, acc, /*reuse_a=*/false, /*reuse_b=*/false);
    aptr += 4;
    bptr += 4 * (size_t)N;
  }

  const float bb = bias ? bias[col] : 0.0f;
  // C/D layout: VGPR j -> rows m0+j (lanes 0-15) / m0+8+j (lanes 16-31)
  const int rbase = m0 + (lane >> 4) * 8;
  #pragma unroll
  for (int j = 0; j < 8; ++j) {
    C[(size_t)(rbase + j) * N + col] = acc[j] + bb;
  }
}

// One block per (n, y) attention row: e[x] = sum_a tanh(ps[a]+ph[x][a])*v[a],
// then numerically-stable softmax over x (Tx=1024, 256 threads x 4 each).
__global__ __launch_bounds__(256)
void attn_tanh_softmax_kernel(const float* __restrict__ ph, const float* __restrict__ ps,
                              const float* __restrict__ v, float* __restrict__ out) {
  __shared__ float sps[CA];
  __shared__ float sv[CA];
  __shared__ float red[256];

  const int tid = threadIdx.x;
  const int row = blockIdx.x;                 // n*Ty + y
  const int n   = row >> 7;                   // Ty = 128

  if (tid < CA) {
    sps[tid] = ps[(size_t)row * CA + tid];
    sv[tid]  = v[tid];
  }
  __syncthreads();

  const float* phn = ph + (size_t)n * TX * CA;
  float e[4];
  #pragma unroll
  for (int i = 0; i < 4; ++i) {
    const int x = tid + i * 256;
    const float4* p4 = (const float4*)(phn + (size_t)x * CA);
    float acc = 0.0f;
    #pragma unroll 4
    for (int a = 0; a < CA / 4; ++a) {
      float4 pv = p4[a];
      acc += hw_tanhf(sps[4 * a + 0] + pv.x) * sv[4 * a + 0];
      acc += hw_tanhf(sps[4 * a + 1] + pv.y) * sv[4 * a + 1];
      acc += hw_tanhf(sps[4 * a + 2] + pv.z) * sv[4 * a + 2];
      acc += hw_tanhf(sps[4 * a + 3] + pv.w) * sv[4 * a + 3];
    }
    e[i] = acc;
  }

  // block-wide max
  float m = fmaxf(fmaxf(e[0], e[1]), fmaxf(e[2], e[3]));
  red[tid] = m;
  __syncthreads();
  for (int s2 = 128; s2 > 0; s2 >>= 1) {
    if (tid < s2) red[tid] = fmaxf(red[tid], red[tid + s2]);
    __syncthreads();
  }
  const float rowmax = red[0];
  __syncthreads();

  // exp + block-wide sum
  float sum = 0.0f;
  #pragma unroll
  for (int i = 0; i < 4; ++i) { e[i] = __expf(e[i] - rowmax); sum += e[i]; }
  red[tid] = sum;
  __syncthreads();
  for (int s2 = 128; s2 > 0; s2 >>= 1) {
    if (tid < s2) red[tid] += red[tid + s2];
    __syncthreads();
  }
  const float inv = 1.0f / red[0];

  #pragma unroll
  for (int i = 0; i < 4; ++i) {
    out[(size_t)row * TX + tid + i * 256] = e[i] * inv;
  }
}

extern "C" void kernel_launch(void* const* d_in, const int* in_sizes, int n_in,
                              void* d_out, int out_size, void* d_ws, size_t ws_size,
                              hipStream_t stream) {
  (void)in_sizes; (void)n_in; (void)out_size; (void)ws_size;

  const float* h  = (const float*)d_in[0];   // (8, 1024, 512)
  const float* s  = (const float*)d_in[1];   // (8, 128, 1024)
  const float* Wh = (const float*)d_in[2];   // (512, 128)
  const float* Ws = (const float*)d_in[3];   // (1024, 128)
  const float* b  = (const float*)d_in[4];   // (128,)
  const float* vv = (const float*)d_in[5];   // (128,)
  float* out = (float*)d_out;                // (8, 128, 1024)

  float* ph = (float*)d_ws;                  // 8*1024*128 floats = 4 MB
  float* ps = ph + (size_t)BN * TX * CA;     // 8*128*128 floats = 0.5 MB

  // ph = h @ Wh              : M = 8192, K = 512, N = 128
  gemm_wmma_f32_kernel<<<(BN * TX) / 16, 256, 0, stream>>>(h, Wh, nullptr, ph,
                                                           BN * TX, ENC, CA);
  // ps = s @ Ws + b          : M = 1024, K = 1024, N = 128
  gemm_wmma_f32_kernel<<<(BN * TY) / 16, 256, 0, stream>>>(s, Ws, b, ps,
                                                           BN * TY, DEC, CA);
  // e -> softmax             : one block per (n, y) row
  attn_tanh_softmax_kernel<<<BN * TY, 256, 0, stream>>>(ph, ps, vv, out);
}